// GAT_v2_ActorCritic_77111842832601
// MI455X (gfx1250) — compile-verified
//
#include <hip/hip_runtime.h>
#include <hip/hip_bf16.h>

// ---------------------------------------------------------------------------
// Problem constants (from reference)
// ---------------------------------------------------------------------------
constexpr int cN  = 50000;
constexpr int cE  = 800000;
constexpr int cB  = 16;
constexpr int cH  = 4;       // heads
constexpr int cF  = 128;     // H*C feature width
constexpr int cET = cE + cN; // edges + self loops

typedef float v2f __attribute__((ext_vector_type(2)));
typedef float v8f __attribute__((ext_vector_type(8)));

// ---------------------------------------------------------------------------
// Helpers
// ---------------------------------------------------------------------------
__device__ __forceinline__ unsigned fenc(float f) {
    unsigned u = __float_as_uint(f);
    return (u & 0x80000000u) ? ~u : (u | 0x80000000u);
}
__device__ __forceinline__ float fdec(unsigned e) {
    return (e & 0x80000000u) ? __uint_as_float(e & 0x7FFFFFFFu)
                             : __uint_as_float(~e);
}
__device__ __forceinline__ void atomAdd(float* p, float v) {
    unsafeAtomicAdd(p, v);   // hardware global_atomic_add_f32 on gfx1250
}

// ---------------------------------------------------------------------------
// Generic zero fill
// ---------------------------------------------------------------------------
__global__ void __launch_bounds__(256) fill_zero(float* __restrict__ p, long long n) {
    long long i = (long long)blockIdx.x * blockDim.x + threadIdx.x;
    if (i < n) p[i] = 0.0f;
}

// ---------------------------------------------------------------------------
// Self-loop edge-attr: per-target mean of incoming edge_attr
// ---------------------------------------------------------------------------
__global__ void __launch_bounds__(256) selfloop_accum(
    const int* __restrict__ ei, const float* __restrict__ ea,
    float* __restrict__ cnt, float* __restrict__ ea_loop) {
    int e = blockIdx.x * blockDim.x + threadIdx.x;
    if (e >= cE) return;
    int dst = ei[cE + e];
    atomAdd(&cnt[dst], 1.0f);
    atomAdd(&ea_loop[2 * dst + 0], ea[2 * e + 0]);
    atomAdd(&ea_loop[2 * dst + 1], ea[2 * e + 1]);
}
__global__ void __launch_bounds__(256) selfloop_div(
    const float* __restrict__ cnt, float* __restrict__ ea_loop) {
    int n = blockIdx.x * blockDim.x + threadIdx.x;
    if (n >= cN) return;
    float c = fmaxf(cnt[n], 1.0f);
    ea_loop[2 * n + 0] /= c;
    ea_loop[2 * n + 1] /= c;
}

// ---------------------------------------------------------------------------
// conv1 projections: x(N x 2) @ W(2 x 128)
// ---------------------------------------------------------------------------
__global__ void __launch_bounds__(256) proj2(
    const float* __restrict__ x, const float* __restrict__ W,
    float* __restrict__ out) {
    int tid = blockIdx.x * blockDim.x + threadIdx.x;
    if (tid >= cN * cF) return;
    int n = tid >> 7, j = tid & 127;
    out[tid] = x[2 * n] * W[j] + x[2 * n + 1] * W[cF + j];
}

// ---------------------------------------------------------------------------
// GATv2 edge phases
// ---------------------------------------------------------------------------
// Phase A: per (edge, head) attention logit + segment max.
// xl/xr gathers are 128-B aligned float4 (global_load_b128); We/att staged in LDS.
__global__ void __launch_bounds__(256) phaseA(
    const float* __restrict__ xl, const float* __restrict__ xr,
    const int* __restrict__ ei, const float* __restrict__ ea,
    const float* __restrict__ ea_loop, const float* __restrict__ We,
    const float* __restrict__ att, float* __restrict__ logits,
    unsigned* __restrict__ mEnc) {
    __shared__ __align__(16) float sW0[cF];
    __shared__ __align__(16) float sW1[cF];
    __shared__ __align__(16) float sAt[cF];
    if (threadIdx.x < cF) {
        sW0[threadIdx.x] = We[threadIdx.x];
        sW1[threadIdx.x] = We[cF + threadIdx.x];
        sAt[threadIdx.x] = att[threadIdx.x];
    }
    __syncthreads();

    int tid = blockIdx.x * blockDim.x + threadIdx.x;
    if (tid >= cET * cH) return;
    int e = tid >> 2, h = tid & 3;
    int src, dst; float a0, a1;
    if (e < cE) {
        src = ei[e]; dst = ei[cE + e];
        a0 = ea[2 * e]; a1 = ea[2 * e + 1];
    } else {
        src = dst = e - cE;
        a0 = ea_loop[2 * (e - cE)]; a1 = ea_loop[2 * (e - cE) + 1];
    }
    const float4* pl = (const float4*)(xl + (long long)src * cF + h * 32);
    const float4* pr = (const float4*)(xr + (long long)dst * cF + h * 32);
    const float4* w0 = (const float4*)(sW0 + h * 32);
    const float4* w1 = (const float4*)(sW1 + h * 32);
    const float4* at = (const float4*)(sAt + h * 32);
    float s = 0.0f;
#pragma unroll
    for (int q = 0; q < 8; ++q) {
        float4 l = pl[q], r = pr[q], e0 = w0[q], e1 = w1[q], a = at[q];
        float v;
        v = l.x + r.x + a0 * e0.x + a1 * e1.x; v = (v > 0.f) ? v : 0.2f * v; s += a.x * v;
        v = l.y + r.y + a0 * e0.y + a1 * e1.y; v = (v > 0.f) ? v : 0.2f * v; s += a.y * v;
        v = l.z + r.z + a0 * e0.z + a1 * e1.z; v = (v > 0.f) ? v : 0.2f * v; s += a.z * v;
        v = l.w + r.w + a0 * e0.w + a1 * e1.w; v = (v > 0.f) ? v : 0.2f * v; s += a.w * v;
    }
    logits[tid] = s;
    atomicMax(&mEnc[dst * cH + h], fenc(s));
}

__global__ void __launch_bounds__(256) phaseB(
    const int* __restrict__ ei, float* __restrict__ logits,
    const unsigned* __restrict__ mEnc, float* __restrict__ den) {
    int tid = blockIdx.x * blockDim.x + threadIdx.x;
    if (tid >= cET * cH) return;
    int e = tid >> 2, h = tid & 3;
    int dst = (e < cE) ? ei[cE + e] : (e - cE);
    float m  = fdec(mEnc[dst * cH + h]);
    float ex = expf(logits[tid] - m);
    logits[tid] = ex;                            // reuse buffer for ex
    atomAdd(&den[dst * cH + h], ex);
}

// Phase C: one thread per (edge, 4 channels): float4 gather + 4 HW f32 atomics.
__global__ void __launch_bounds__(256) phaseC(
    const float* __restrict__ xl, const int* __restrict__ ei,
    const float* __restrict__ ex, const float* __restrict__ den,
    float* __restrict__ out) {
    int tid = blockIdx.x * blockDim.x + threadIdx.x;
    if (tid >= cET * 32) return;
    int e = tid >> 5, q = tid & 31;
    int j = q * 4, h = q >> 3;
    int src, dst;
    if (e < cE) { src = ei[e]; dst = ei[cE + e]; }
    else        { src = dst = e - cE; }
    float alpha = ex[e * cH + h] / (den[dst * cH + h] + 1e-16f);
    float4 v = *(const float4*)(xl + (long long)src * cF + j);
    float* po = out + (long long)dst * cF + j;
    atomAdd(po + 0, alpha * v.x);
    atomAdd(po + 1, alpha * v.y);
    atomAdd(po + 2, alpha * v.z);
    atomAdd(po + 3, alpha * v.w);
}

__global__ void __launch_bounds__(256) bias_tanh(
    float* __restrict__ y, const float* __restrict__ bias) {
    int tid = blockIdx.x * blockDim.x + threadIdx.x;
    if (tid >= cN * cF) return;
    y[tid] = tanhf(y[tid] + bias[tid & 127]);
}

// ---------------------------------------------------------------------------
// WMMA dual GEMM: O0 = A @ W0, O1 = A @ W1  (A: 50000x128, W: 128x128, f32)
// All 8 waves of a block share one 16-row A strip (tm = blockIdx.x, tn = wave),
// so the strip (16x128 f32 = 8 KB, contiguous) is staged in LDS once and
// fragments are re-read via ds_load.  K-loop uses V_WMMA_F32_16X16X4_F32.
// A 16x4 layout : lanes 0-15 hold K0(V0),K1(V1); lanes 16-31 hold K2,K3.
// B  4x16 layout: lanes 0-15 hold rows K0(V0),K1(V1); lanes 16-31 rows K2,K3.
// C/D 16x16    : VGPR r -> rows r (lanes 0-15) and r+8 (lanes 16-31).
// ---------------------------------------------------------------------------
__global__ void __launch_bounds__(256) wmma_dual_gemm(
    const float* __restrict__ A, const float* __restrict__ W0,
    const float* __restrict__ W1, float* __restrict__ O0,
    float* __restrict__ O1) {
    __shared__ __align__(16) float sA[16 * cF];   // 8 KB strip
    int tm = blockIdx.x;                          // 3125 blocks
    const float* Ab = A + (long long)tm * 16 * cF;
    for (int i = threadIdx.x * 4; i < 16 * cF; i += 256 * 4)
        *(float4*)(sA + i) = *(const float4*)(Ab + i);
    __syncthreads();

    int tn   = threadIdx.x >> 5;                  // wave id = output col tile
    int lane = threadIdx.x & 31;
    int row = lane & 15;
    int kb  = (lane >> 4) * 2;                    // 0 or 2
    int col = lane & 15;

    v8f acc0 = {}; v8f acc1 = {};
    const float* arow = sA + row * cF;

    for (int k0 = 0; k0 < cF; k0 += 4) {
        v2f a;
        a.x = arow[k0 + kb];
        a.y = arow[k0 + kb + 1];
        v2f b0, b1;
        b0.x = W0[(k0 + kb)     * cF + tn * 16 + col];
        b0.y = W0[(k0 + kb + 1) * cF + tn * 16 + col];
        b1.x = W1[(k0 + kb)     * cF + tn * 16 + col];
        b1.y = W1[(k0 + kb + 1) * cF + tn * 16 + col];
        acc0 = __builtin_amdgcn_wmma_f32_16x16x4_f32(
            false, a, false, b0, (short)0, acc0, false, false);
        acc1 = __builtin_amdgcn_wmma_f32_16x16x4_f32(
            false, a, false, b1, (short)0, acc1, false, false);
    }
    int rbase = (lane >> 4) * 8;
    for (int r = 0; r < 8; ++r) {
        long long o = (long long)(tm * 16 + rbase + r) * cF + tn * 16 + col;
        O0[o] = acc0[r];
        O1[o] = acc1[r];
    }
}

// ---------------------------------------------------------------------------
// Global mean pool (float4 loads + HW f32 atomics)
// ---------------------------------------------------------------------------
__global__ void __launch_bounds__(256) pool_accum(
    const float* __restrict__ y, const int* __restrict__ batch,
    float* __restrict__ g, float* __restrict__ gcnt) {
    int tid = blockIdx.x * blockDim.x + threadIdx.x;
    if (tid >= cN * 32) return;
    int n = tid >> 5, j = (tid & 31) * 4;
    int b = batch[n];
    float4 v = *(const float4*)(y + (long long)n * cF + j);
    float* pg = g + b * cF + j;
    atomAdd(pg + 0, v.x);
    atomAdd(pg + 1, v.y);
    atomAdd(pg + 2, v.z);
    atomAdd(pg + 3, v.w);
    if ((tid & 31) == 0) atomAdd(&gcnt[b], 1.0f);
}
__global__ void __launch_bounds__(256) pool_div(
    float* __restrict__ g, const float* __restrict__ gcnt) {
    int tid = blockIdx.x * blockDim.x + threadIdx.x;
    if (tid >= cB * cF) return;
    g[tid] /= fmaxf(gcnt[tid >> 7], 1.0f);
}

// ---------------------------------------------------------------------------
// MLP head blocks: Linear -> LayerNorm -> tanh  (one block per batch row)
// ---------------------------------------------------------------------------
__global__ void __launch_bounds__(256) lin_ln_tanh(
    const float* __restrict__ in, const float* __restrict__ W,
    const float* __restrict__ b, const float* __restrict__ gmm,
    const float* __restrict__ bet, float* __restrict__ out,
    int din, int dout) {
    __shared__ float h[512];
    __shared__ float s1[256];
    __shared__ float s2[256];
    int rowi = blockIdx.x;
    const float* xin = in + (long long)rowi * din;
    for (int j = threadIdx.x; j < dout; j += blockDim.x) {
        float s = b[j];
        for (int k = 0; k < din; ++k) s += xin[k] * W[k * dout + j];
        h[j] = s;
    }
    __syncthreads();
    float ls = 0.0f, lq = 0.0f;
    for (int j = threadIdx.x; j < dout; j += blockDim.x) {
        float v = h[j];
        ls += v; lq += v * v;
    }
    s1[threadIdx.x] = ls; s2[threadIdx.x] = lq;
    __syncthreads();
    for (int st = 128; st > 0; st >>= 1) {
        if (threadIdx.x < st) {
            s1[threadIdx.x] += s1[threadIdx.x + st];
            s2[threadIdx.x] += s2[threadIdx.x + st];
        }
        __syncthreads();
    }
    float mu  = s1[0] / (float)dout;
    float var = s2[0] / (float)dout - mu * mu;
    float inv = rsqrtf(var + 1e-5f);
    for (int j = threadIdx.x; j < dout; j += blockDim.x)
        out[(long long)rowi * dout + j] = tanhf((h[j] - mu) * inv * gmm[j] + bet[j]);
}

// Final linear; mode 0 = identity, 1 = sigmoid, 2 = tanh*0.693-2.303
__global__ void __launch_bounds__(32) lin_out(
    const float* __restrict__ in, const float* __restrict__ W,
    const float* __restrict__ b, float* __restrict__ out,
    int din, int dout, int mode) {
    int rowi = blockIdx.x;
    int j = threadIdx.x;
    if (j >= dout) return;
    const float* xin = in + (long long)rowi * din;
    float s = b[j];
    for (int k = 0; k < din; ++k) s += xin[k] * W[k * dout + j];
    if (mode == 1)      s = 1.0f / (1.0f + expf(-s));
    else if (mode == 2) s = tanhf(s) * 0.693f - 2.303f;
    out[(long long)rowi * dout + j] = s;
}

// ---------------------------------------------------------------------------
// Host launcher
// ---------------------------------------------------------------------------
extern "C" void kernel_launch(void* const* d_in, const int* in_sizes, int n_in,
                              void* d_out, int out_size, void* d_ws, size_t ws_size,
                              hipStream_t stream) {
    (void)in_sizes; (void)n_in; (void)out_size; (void)ws_size;

    const float* x     = (const float*)d_in[0];
    const int*   ei    = (const int*)  d_in[1];
    const float* ea    = (const float*)d_in[2];
    const int*   batch = (const int*)  d_in[3];
    auto P = [&](int i) { return (const float*)d_in[i]; };
    // params pytree flatten order (sorted dict keys, W<a<b):
    // 4:c1.We 5:c1.Wl 6:c1.Wr 7:c1.att 8:c1.bias
    // 9:c2.We 10:c2.Wl 11:c2.Wr 12:c2.att 13:c2.bias
    // 14-25:logstd_mlp  26,27:logstd_out
    // 28-39:means_mlp   40,41:means_out
    // 42-53:mix_mlp     54,55:mix_out
    // 56,57:np_out      58-69:sample_mlp   70-77:shared

    float* ws = (float*)d_ws;
    size_t o = 0;
    float*    cnt      = ws + o; o += cN;
    float*    ea_loop  = ws + o; o += 2LL * cN;
    float*    bufXL    = ws + o; o += (size_t)cN * cF;
    float*    bufXR    = ws + o; o += (size_t)cN * cF;
    float*    bufY     = ws + o; o += (size_t)cN * cF;
    float*    logitsEx = ws + o; o += (size_t)cET * cH;
    unsigned* mEnc     = (unsigned*)(ws + o); o += (size_t)cN * cH;
    float*    den      = ws + o; o += (size_t)cN * cH;      // adjacent to mEnc
    float*    g        = ws + o; o += (size_t)cB * cF;
    float*    gcnt     = ws + o; o += cB;                   // adjacent to g
    float*    h0       = ws + o; o += (size_t)cB * 512;
    float*    h1       = ws + o; o += (size_t)cB * 512;
    float*    h2       = ws + o; o += (size_t)cB * 256;

    float* outMeans  = (float*)d_out;        // 16*6
    float* outLogStd = (float*)d_out + 96;   // 16*6
    float* outMix    = (float*)d_out + 192;  // 16*3
    float* outNp     = (float*)d_out + 240;  // 16*10

    auto nb = [](long long n) { return dim3((unsigned)((n + 255) / 256)); };
    const dim3 T(256);
    const long long nETH  = (long long)cET * cH;
    const long long nET32 = (long long)cET * 32;
    const long long nNF   = (long long)cN  * cF;
    const long long nN32  = (long long)cN  * 32;

    // --- self loop edge attr (mean of incoming) ---
    fill_zero<<<nb(3LL * cN), T, 0, stream>>>(cnt, 3LL * cN); // cnt + ea_loop
    selfloop_accum<<<nb(cE), T, 0, stream>>>(ei, ea, cnt, ea_loop);
    selfloop_div<<<nb(cN), T, 0, stream>>>(cnt, ea_loop);

    // --- conv1 projections (K=2, plain) ---
    proj2<<<nb(nNF), T, 0, stream>>>(x, P(5), bufXL);   // Wl1
    proj2<<<nb(nNF), T, 0, stream>>>(x, P(6), bufXR);   // Wr1

    auto run_conv = [&](const float* We, const float* att, const float* bias) {
        fill_zero<<<nb(2LL * cN * cH), T, 0, stream>>>((float*)mEnc, 2LL * cN * cH);
        fill_zero<<<nb(nNF), T, 0, stream>>>(bufY, nNF);
        phaseA<<<nb(nETH), T, 0, stream>>>(bufXL, bufXR, ei, ea, ea_loop,
                                           We, att, logitsEx, mEnc);
        phaseB<<<nb(nETH), T, 0, stream>>>(ei, logitsEx, mEnc, den);
        phaseC<<<nb(nET32), T, 0, stream>>>(bufXL, ei, logitsEx, den, bufY);
        bias_tanh<<<nb(nNF), T, 0, stream>>>(bufY, bias);
    };

    // --- conv1 ---
    run_conv(P(4), P(7), P(8));

    // --- conv2 projections via WMMA (y1 @ Wl2 , y1 @ Wr2) ---
    wmma_dual_gemm<<<dim3(cN / 16), T, 0, stream>>>(bufY, P(10), P(11), bufXL, bufXR);

    // --- conv2 ---
    run_conv(P(9), P(12), P(13));

    // --- global mean pool ---
    fill_zero<<<nb(cB * cF + cB), T, 0, stream>>>(g, (long long)cB * cF + cB);
    pool_accum<<<nb(nN32), T, 0, stream>>>(bufY, batch, g, gcnt);
    pool_div<<<nb(cB * cF), T, 0, stream>>>(g, gcnt);

    // --- MLP heads (B=16 rows) ---
    // shared: 128->512->256
    lin_ln_tanh<<<cB, T, 0, stream>>>(g,  P(70), P(71), P(72), P(73), h0, 128, 512);
    lin_ln_tanh<<<cB, T, 0, stream>>>(h0, P(74), P(75), P(76), P(77), h1, 512, 256);
    // means: 256->256->128->64 -> 6 (sigmoid)
    lin_ln_tanh<<<cB, T, 0, stream>>>(h1, P(28), P(29), P(30), P(31), h0, 256, 256);
    lin_ln_tanh<<<cB, T, 0, stream>>>(h0, P(32), P(33), P(34), P(35), h2, 256, 128);
    lin_ln_tanh<<<cB, T, 0, stream>>>(h2, P(36), P(37), P(38), P(39), h0, 128, 64);
    lin_out<<<cB, 32, 0, stream>>>(h0, P(40), P(41), outMeans, 64, 6, 1);
    // log_stds: 256->256->128->64 -> 6 (tanh*0.693-2.303)
    lin_ln_tanh<<<cB, T, 0, stream>>>(h1, P(14), P(15), P(16), P(17), h0, 256, 256);
    lin_ln_tanh<<<cB, T, 0, stream>>>(h0, P(18), P(19), P(20), P(21), h2, 256, 128);
    lin_ln_tanh<<<cB, T, 0, stream>>>(h2, P(22), P(23), P(24), P(25), h0, 128, 64);
    lin_out<<<cB, 32, 0, stream>>>(h0, P(26), P(27), outLogStd, 64, 6, 2);
    // mix: 256->256->128->64 -> 3
    lin_ln_tanh<<<cB, T, 0, stream>>>(h1, P(42), P(43), P(44), P(45), h0, 256, 256);
    lin_ln_tanh<<<cB, T, 0, stream>>>(h0, P(46), P(47), P(48), P(49), h2, 256, 128);
    lin_ln_tanh<<<cB, T, 0, stream>>>(h2, P(50), P(51), P(52), P(53), h0, 128, 64);
    lin_out<<<cB, 32, 0, stream>>>(h0, P(54), P(55), outMix, 64, 3, 0);
    // sample: 128->256->128->64 -> 10
    lin_ln_tanh<<<cB, T, 0, stream>>>(g,  P(58), P(59), P(60), P(61), h0, 128, 256);
    lin_ln_tanh<<<cB, T, 0, stream>>>(h0, P(62), P(63), P(64), P(65), h2, 256, 128);
    lin_ln_tanh<<<cB, T, 0, stream>>>(h2, P(66), P(67), P(68), P(69), h0, 128, 64);
    lin_out<<<cB, 32, 0, stream>>>(h0, P(56), P(57), outNp, 64, 10, 0);
}